// S4Base_5102421148272
// MI455X (gfx1250) — compile-verified
//
#include <hip/hip_runtime.h>
#include <hip/hip_bf16.h>
#include <stdint.h>

// ---------------------------------------------------------------------------
// S4 (DPLR) forward: kernel generation via Cauchy GEMM (WMMA f32 16x16x4)
// + mixed-radix Stockham FFT convolution (radix-8 main passes) with fused
// pad / spectral-multiply / epilogue passes.  ~12GB HBM traffic (~1ms).
// ---------------------------------------------------------------------------

#define Bsz   4
#define Llen  16384
#define Dch   512
#define Nst   64
#define NFFT  32768
#define PI_F  3.14159265358979323846f

typedef __attribute__((ext_vector_type(2))) float v2f;
typedef __attribute__((ext_vector_type(8))) float v8f;

#ifndef __has_builtin
#define __has_builtin(x) 0
#endif
#if __has_builtin(__builtin_amdgcn_wmma_f32_16x16x4_f32)
#define S4_WMMA 1
#else
#define S4_WMMA 0
#endif

// ------------------------- complex helpers ---------------------------------
__device__ __forceinline__ float2 cadd(float2 a, float2 b) { return make_float2(a.x + b.x, a.y + b.y); }
__device__ __forceinline__ float2 csub(float2 a, float2 b) { return make_float2(a.x - b.x, a.y - b.y); }
__device__ __forceinline__ float2 cmulf(float2 a, float2 b) { return make_float2(a.x * b.x - a.y * b.y, a.x * b.y + a.y * b.x); }
__device__ __forceinline__ float2 crotis(float2 a, float s) { return make_float2(-s * a.y, s * a.x); }  // s*i*a

// 8-point DFT (DIF), X[m] = sum_r v[r] * e^{s*2*pi*i*r*m/8}, natural order out.
__device__ __forceinline__ void dft8(const float2* v, float s, float2* X) {
  const float R = 0.70710678118654752440f;
  float2 t0 = cadd(v[0], v[4]), t1 = cadd(v[1], v[5]);
  float2 t2 = cadd(v[2], v[6]), t3 = cadd(v[3], v[7]);
  float2 u0 = csub(v[0], v[4]), u1 = csub(v[1], v[5]);
  float2 u2 = csub(v[2], v[6]), u3 = csub(v[3], v[7]);
  float2 m1 = cmulf(u1, make_float2(R, s * R));      // w8^1
  float2 m2 = crotis(u2, s);                         // w8^2 = s*i
  float2 m3 = cmulf(u3, make_float2(-R, s * R));     // w8^3
  // even outputs: DFT4(t)
  float2 p0 = cadd(t0, t2), p1 = csub(t0, t2);
  float2 p2 = cadd(t1, t3), p3 = crotis(csub(t1, t3), s);
  X[0] = cadd(p0, p2); X[2] = cadd(p1, p3);
  X[4] = csub(p0, p2); X[6] = csub(p1, p3);
  // odd outputs: DFT4(u0, m1, m2, m3)
  float2 q0 = cadd(u0, m2), q1 = csub(u0, m2);
  float2 q2 = cadd(m1, m3), q3 = crotis(csub(m1, m3), s);
  X[1] = cadd(q0, q2); X[3] = cadd(q1, q3);
  X[5] = csub(q0, q2); X[7] = csub(q1, q3);
}

// ---------------------------------------------------------------------------
// Per-l scalars: Omega, g, scale = 2/(1+Omega), s = k01/(1+k11).
// perl layout: [0,L) sr | [L,2L) si | [2L,3L) scale_r | [3L,4L) scale_i
//              [4L,5L) g_r | [5L,6L) g_i
// ---------------------------------------------------------------------------
__global__ __launch_bounds__(256) void s4_perl(
    const float* __restrict__ Lam, const float* __restrict__ Pri,
    const float* __restrict__ Cri, const float* __restrict__ log_step,
    float* __restrict__ perl)
{
  int l = blockIdx.x * blockDim.x + threadIdx.x;
  if (l >= Llen) return;
  float step = expf(log_step[0]);
  float theta = -2.0f * PI_F * (float)l / (float)Llen;
  float oi, orr;
  sincosf(theta, &oi, &orr);
  float op1r = 1.0f + orr;
  float den = op1r * op1r + oi * oi;
  float nr = 1.0f - orr, ni = -oi;
  float gscale = 2.0f / step;
  float gr = gscale * (nr * op1r + ni * oi) / den;
  float gi = gscale * (ni * op1r - nr * oi) / den;
  float scr = 2.0f * op1r / den;
  float sci = -2.0f * oi / den;
  float k01r = 0.f, k01i = 0.f, k11r = 0.f, k11i = 0.f;
  for (int n = 0; n < Nst; ++n) {
    float lr = Lam[2 * n], li = Lam[2 * n + 1];
    float pr = Pri[2 * n], pi = Pri[2 * n + 1];
    float cr = Cri[2 * n], ci = Cri[2 * n + 1];
    float zr = gr - lr, zi = gi - li;
    float zd = zr * zr + zi * zi;
    float car = zr / zd, cai = -zi / zd;
    float w01r = cr * pr + ci * pi;
    float w01i = cr * pi - ci * pr;
    float w11r = pr * pr + pi * pi;
    k01r += car * w01r - cai * w01i;
    k01i += car * w01i + cai * w01r;
    k11r += car * w11r;
    k11i += cai * w11r;
  }
  float dr = 1.0f + k11r, di = k11i;
  float dd = dr * dr + di * di;
  float sr = (k01r * dr + k01i * di) / dd;
  float si = (k01i * dr - k01r * di) / dd;
  perl[l] = sr;
  perl[Llen + l] = si;
  perl[2 * Llen + l] = scr;
  perl[3 * Llen + l] = sci;
  perl[4 * Llen + l] = gr;
  perl[5 * Llen + l] = gi;
}

// ---------------------------------------------------------------------------
// Weight matrices W0 = conj(C).*B, W1 = conj(P).*B, K-pair packed:
// W[(k>>1)*2D + 2d + (k&1)] so a WMMA B-fragment is one float2 load.
// ---------------------------------------------------------------------------
__global__ __launch_bounds__(256) void s4_weights(
    const float* __restrict__ Bri, const float* __restrict__ Pri,
    const float* __restrict__ Cri,
    float* __restrict__ W0r, float* __restrict__ W0i,
    float* __restrict__ W1r, float* __restrict__ W1i)
{
  int idx = blockIdx.x * blockDim.x + threadIdx.x;
  if (idx >= Nst * Dch) return;
  int n = idx / Dch, d = idx % Dch;
  float br = Bri[2 * (n * Dch + d)], bi = Bri[2 * (n * Dch + d) + 1];
  float cr = Cri[2 * n], ci = Cri[2 * n + 1];
  float pr = Pri[2 * n], pi = Pri[2 * n + 1];
  int o = (n >> 1) * (2 * Dch) + 2 * d + (n & 1);
  W0r[o] = cr * br + ci * bi;
  W0i[o] = cr * bi - ci * br;
  W1r[o] = pr * br + pi * bi;
  W1i[o] = pr * bi - pi * br;
}

// ---------------------------------------------------------------------------
// Cauchy matrix A[l,n] = 1/(g_l - lambda_n), K-pair packed along l.
// ---------------------------------------------------------------------------
__global__ __launch_bounds__(256) void s4_cauchy(
    const float* __restrict__ Lam, const float* __restrict__ perl,
    float* __restrict__ APr, float* __restrict__ APi)
{
  int idx = blockIdx.x * blockDim.x + threadIdx.x;
  if (idx >= Llen * Nst) return;
  int n = idx / Llen, l = idx % Llen;
  float gr = perl[4 * Llen + l], gi = perl[5 * Llen + l];
  float zr = gr - Lam[2 * n], zi = gi - Lam[2 * n + 1];
  float zd = zr * zr + zi * zi;
  int o = (n >> 1) * (2 * Llen) + 2 * l + (n & 1);
  APr[o] = zr / zd;
  APi[o] = -zi / zd;
}

// ---------------------------------------------------------------------------
// Complex GEMM (L x 64) @ (64 x 512) for k00/k10 via WMMA f32 16x16x4,
// fused with rank-1 correction:  Ev[l,d] = scale_l * (k00 - s_l * k10)
// ---------------------------------------------------------------------------
__global__ __launch_bounds__(32) void s4_cauchy_gemm(
    const float* __restrict__ APr, const float* __restrict__ APi,
    const float* __restrict__ W0r, const float* __restrict__ W0i,
    const float* __restrict__ W1r, const float* __restrict__ W1i,
    const float* __restrict__ perl, float2* __restrict__ Ev)
{
  int lane = threadIdx.x & 31;
  int tileD = blockIdx.x;
  int tileL = blockIdx.y;
  int d0 = tileD * 16 + (lane & 15);
  int mbase = tileL * 16 + ((lane >> 4) ? 8 : 0);
#if S4_WMMA
  int m = tileL * 16 + (lane & 15);
  int khi = (lane >> 4) * 2;
  v8f aRr = {}, aRi = {}, aSr = {}, aSi = {};
  for (int kb = 0; kb < Nst; kb += 4) {
    int k0 = kb + khi;
    const float* pr = APr + (k0 >> 1) * (2 * Llen) + 2 * m;
    const float* pi = APi + (k0 >> 1) * (2 * Llen) + 2 * m;
    v2f ar = *(const v2f*)pr;
    v2f ai = *(const v2f*)pi;
    v2f nai = -ai;
    int bo = (k0 >> 1) * (2 * Dch) + 2 * d0;
    v2f b0r = *(const v2f*)(W0r + bo);
    v2f b0i = *(const v2f*)(W0i + bo);
    v2f b1r = *(const v2f*)(W1r + bo);
    v2f b1i = *(const v2f*)(W1i + bo);
    aRr = __builtin_amdgcn_wmma_f32_16x16x4_f32(false, ar,  false, b0r, (short)0, aRr, false, false);
    aRr = __builtin_amdgcn_wmma_f32_16x16x4_f32(false, nai, false, b0i, (short)0, aRr, false, false);
    aRi = __builtin_amdgcn_wmma_f32_16x16x4_f32(false, ar,  false, b0i, (short)0, aRi, false, false);
    aRi = __builtin_amdgcn_wmma_f32_16x16x4_f32(false, ai,  false, b0r, (short)0, aRi, false, false);
    aSr = __builtin_amdgcn_wmma_f32_16x16x4_f32(false, ar,  false, b1r, (short)0, aSr, false, false);
    aSr = __builtin_amdgcn_wmma_f32_16x16x4_f32(false, nai, false, b1i, (short)0, aSr, false, false);
    aSi = __builtin_amdgcn_wmma_f32_16x16x4_f32(false, ar,  false, b1i, (short)0, aSi, false, false);
    aSi = __builtin_amdgcn_wmma_f32_16x16x4_f32(false, ai,  false, b1r, (short)0, aSi, false, false);
  }
  for (int r = 0; r < 8; ++r) {
    int l = mbase + r;
    float k00r = aRr[r], k00i = aRi[r];
    float k10r = aSr[r], k10i = aSi[r];
    float sr = perl[l],            si = perl[Llen + l];
    float cr = perl[2 * Llen + l], ci = perl[3 * Llen + l];
    float tr = k00r - (sr * k10r - si * k10i);
    float ti = k00i - (sr * k10i + si * k10r);
    Ev[l * Dch + d0] = make_float2(cr * tr - ci * ti, cr * ti + ci * tr);
  }
#else
  for (int r = 0; r < 8; ++r) {
    int l = mbase + r;
    float k00r = 0.f, k00i = 0.f, k10r = 0.f, k10i = 0.f;
    for (int k = 0; k < Nst; ++k) {
      int ao = (k >> 1) * (2 * Llen) + 2 * l + (k & 1);
      int bo = (k >> 1) * (2 * Dch) + 2 * d0 + (k & 1);
      float arr = APr[ao], aii = APi[ao];
      float b0r = W0r[bo], b0i = W0i[bo];
      float b1r = W1r[bo], b1i = W1i[bo];
      k00r += arr * b0r - aii * b0i;  k00i += arr * b0i + aii * b0r;
      k10r += arr * b1r - aii * b1i;  k10i += arr * b1i + aii * b1r;
    }
    float sr = perl[l],            si = perl[Llen + l];
    float cr = perl[2 * Llen + l], ci = perl[3 * Llen + l];
    float tr = k00r - (sr * k10r - si * k10i);
    float ti = k00i - (sr * k10i + si * k10r);
    Ev[l * Dch + d0] = make_float2(cr * tr - ci * ti, cr * ti + ci * tr);
  }
#endif
}

// ---------------------------------------------------------------------------
// Generic Stockham radix-4 pass (used once: first pass of the L-point ifft).
// ---------------------------------------------------------------------------
__global__ __launch_bounds__(256) void s4_fft_r4(
    const float2* __restrict__ x, float2* __restrict__ y,
    int n, int Ns, float sign)
{
  int t = blockIdx.x * blockDim.x + threadIdx.x;
  int q = n >> 2;
  if (t >= q * Dch) return;
  int col = t & (Dch - 1);
  int j = t >> 9;
  int k = j & (Ns - 1);
  float ang = sign * (0.5f * PI_F) * (float)k / (float)Ns;
  float s1, c1;
  sincosf(ang, &s1, &c1);
  float c2 = c1 * c1 - s1 * s1, s2 = 2.f * c1 * s1;
  float c3 = c2 * c1 - s2 * s1, s3 = c2 * s1 + s2 * c1;
  float2 a = x[j * Dch + col];
  float2 b = x[(j + q) * Dch + col];
  float2 c = x[(j + 2 * q) * Dch + col];
  float2 d = x[(j + 3 * q) * Dch + col];
  float2 v1 = make_float2(c1 * b.x - s1 * b.y, c1 * b.y + s1 * b.x);
  float2 v2 = make_float2(c2 * c.x - s2 * c.y, c2 * c.y + s2 * c.x);
  float2 v3 = make_float2(c3 * d.x - s3 * d.y, c3 * d.y + s3 * d.x);
  float2 p0 = cadd(a, v2), p1 = csub(a, v2);
  float2 p2 = cadd(v1, v3), p3 = crotis(csub(v1, v3), sign);
  int i0 = 4 * j - 3 * k;
  y[(i0         ) * Dch + col] = cadd(p0, p2);
  y[(i0 +     Ns) * Dch + col] = cadd(p1, p3);
  y[(i0 + 2 * Ns) * Dch + col] = csub(p0, p2);
  y[(i0 + 3 * Ns) * Dch + col] = csub(p1, p3);
}

// ---------------------------------------------------------------------------
// Generic Stockham radix-8 pass. sign = -1 forward, +1 inverse (unscaled).
// ---------------------------------------------------------------------------
__global__ __launch_bounds__(256) void s4_fft_r8(
    const float2* __restrict__ x, float2* __restrict__ y,
    int n, int Ns, float sign)
{
  int t = blockIdx.x * blockDim.x + threadIdx.x;
  int q = n >> 3;
  if (t >= q * Dch) return;
  int col = t & (Dch - 1);
  int j = t >> 9;
  int k = j & (Ns - 1);
  float ang = sign * (0.25f * PI_F) * (float)k / (float)Ns;   // 2*pi*k/(8*Ns)
  float s1, c1;
  sincosf(ang, &s1, &c1);
  float2 w1 = make_float2(c1, s1);
  float2 v[8];
  v[0] = x[j * Dch + col];
  float2 w = w1;
#pragma unroll
  for (int r = 1; r < 8; ++r) {
    v[r] = cmulf(x[(j + r * q) * Dch + col], w);
    w = cmulf(w, w1);
  }
  float2 X[8];
  dft8(v, sign, X);
  int i0 = 8 * j - 7 * k;                                     // (j/Ns)*8Ns + k
#pragma unroll
  for (int m = 0; m < 8; ++m)
    y[(i0 + m * Ns) * Dch + col] = X[m];
}

// Fused zero-pad + first forward radix-8 pass (Ns=1 -> all twiddles 1; the
// upper 4 butterfly inputs are the zero padding).  Reads real u directly.
__global__ __launch_bounds__(256) void s4_upad_r8(
    const float* __restrict__ u, float2* __restrict__ y)
{
  const int Q = NFFT / 8;                    // 4096
  int t = blockIdx.x * blockDim.x + threadIdx.x;
  if (t >= Q * Dch) return;
  int col = t & (Dch - 1);
  int j = t >> 9;
  float2 v[8];
#pragma unroll
  for (int r = 0; r < 4; ++r)
    v[r] = make_float2(u[(j + r * Q) * Dch + col], 0.f);
#pragma unroll
  for (int r = 4; r < 8; ++r)
    v[r] = make_float2(0.f, 0.f);
  float2 X[8];
  dft8(v, -1.0f, X);
  int i0 = 8 * j;
#pragma unroll
  for (int m = 0; m < 8; ++m)
    y[(i0 + m) * Dch + col] = X[m];
}

// Same, sourcing Re(ifft result) * (1/L) for the kernel K.
__global__ __launch_bounds__(256) void s4_kpad_r8(
    const float2* __restrict__ Kt, float2* __restrict__ y)
{
  const int Q = NFFT / 8;
  int t = blockIdx.x * blockDim.x + threadIdx.x;
  if (t >= Q * Dch) return;
  int col = t & (Dch - 1);
  int j = t >> 9;
  float2 v[8];
#pragma unroll
  for (int r = 0; r < 4; ++r)
    v[r] = make_float2(Kt[(j + r * Q) * Dch + col].x * (1.0f / (float)Llen), 0.f);
#pragma unroll
  for (int r = 4; r < 8; ++r)
    v[r] = make_float2(0.f, 0.f);
  float2 X[8];
  dft8(v, -1.0f, X);
  int i0 = 8 * j;
#pragma unroll
  for (int m = 0; m < 8; ++m)
    y[(i0 + m) * Dch + col] = X[m];
}

// Fused spectral multiply + first inverse radix-8 pass (Ns=1, twiddle-free).
__global__ __launch_bounds__(256) void s4_cmul_r8(
    const float2* __restrict__ Xs, const float2* __restrict__ Kd,
    float2* __restrict__ y)
{
  const int Q = NFFT / 8;
  int t = blockIdx.x * blockDim.x + threadIdx.x;
  if (t >= Q * Dch) return;
  int col = t & (Dch - 1);
  int j = t >> 9;
  float2 v[8];
#pragma unroll
  for (int r = 0; r < 8; ++r) {
    int idx = (j + r * Q) * Dch + col;
    v[r] = cmulf(Xs[idx], Kd[idx]);
  }
  float2 X[8];
  dft8(v, +1.0f, X);
  int i0 = 8 * j;
#pragma unroll
  for (int m = 0; m < 8; ++m)
    y[(i0 + m) * Dch + col] = X[m];
}

// Fused last inverse radix-8 pass (Ns=4096) + epilogue.  The 8 outputs land
// at j + m*4096; only m=0..3 are < L and only real parts are needed:
//   out = re(X_m)/NFFT + D*u.
__global__ __launch_bounds__(256) void s4_final_r8(
    const float2* __restrict__ x, const float* __restrict__ u,
    const float* __restrict__ Dp, float* __restrict__ out)
{
  const int Q = NFFT / 8;                    // 4096 = Ns of the last pass
  int t = blockIdx.x * blockDim.x + threadIdx.x;
  if (t >= Q * Dch) return;
  int col = t & (Dch - 1);
  int j = t >> 9;
  float ang = (0.25f * PI_F) * (float)j / (float)Q;   // sign=+1, k=j
  float s1, c1;
  sincosf(ang, &s1, &c1);
  float2 w1 = make_float2(c1, s1);
  float2 v[8];
  v[0] = x[j * Dch + col];
  float2 w = w1;
#pragma unroll
  for (int r = 1; r < 8; ++r) {
    v[r] = cmulf(x[(j + r * Q) * Dch + col], w);
    w = cmulf(w, w1);
  }
  float2 X[8];
  dft8(v, +1.0f, X);
  float dp = Dp[0];
  const float inv = 1.0f / (float)NFFT;
#pragma unroll
  for (int m = 0; m < 4; ++m) {
    int pos = (j + m * Q) * Dch + col;
    out[pos] = X[m].x * inv + dp * u[pos];
  }
}

// ---------------------------------------------------------------------------

extern "C" void kernel_launch(void* const* d_in, const int* in_sizes, int n_in,
                              void* d_out, int out_size, void* d_ws, size_t ws_size,
                              hipStream_t stream) {
  const float* u        = (const float*)d_in[0];
  const float* Lam      = (const float*)d_in[1];
  const float* Pri      = (const float*)d_in[2];
  const float* Bri      = (const float*)d_in[3];
  const float* Cri      = (const float*)d_in[4];
  const float* Dp       = (const float*)d_in[5];
  const float* log_step = (const float*)d_in[6];
  float* out = (float*)d_out;

  uint8_t* base = (uint8_t*)d_ws;
  size_t off = 0;
  auto carve = [&](size_t bytes) -> void* {
    void* p = base + off;
    off += (bytes + 255) & ~(size_t)255;
    return p;
  };
  float*  APr  = (float*)carve((size_t)Llen * Nst * 4);
  float*  APi  = (float*)carve((size_t)Llen * Nst * 4);
  float*  W0r  = (float*)carve((size_t)Nst * Dch * 4);
  float*  W0i  = (float*)carve((size_t)Nst * Dch * 4);
  float*  W1r  = (float*)carve((size_t)Nst * Dch * 4);
  float*  W1i  = (float*)carve((size_t)Nst * Dch * 4);
  float*  perl = (float*)carve((size_t)6 * Llen * 4);
  float2* Ev   = (float2*)carve((size_t)Llen * Dch * 8);
  float2* T0   = (float2*)carve((size_t)NFFT * Dch * 8);
  float2* T1   = (float2*)carve((size_t)NFFT * Dch * 8);
  float2* Kd   = (float2*)carve((size_t)NFFT * Dch * 8);
  (void)ws_size; (void)in_sizes; (void)n_in; (void)out_size;

  const int TB   = 256;
  const int gR8  = (NFFT / 8) * Dch / TB;  // 8192 blocks (radix-8, n=32768)
  const int gR8s = (Llen / 8) * Dch / TB;  // 4096 blocks (radix-8, n=16384)
  const int gR4s = (Llen / 4) * Dch / TB;  // 8192 blocks (radix-4, n=16384)

  // --- kernel generation -------------------------------------------------
  s4_perl<<<(Llen + TB - 1) / TB, TB, 0, stream>>>(Lam, Pri, Cri, log_step, perl);
  s4_weights<<<(Nst * Dch + TB - 1) / TB, TB, 0, stream>>>(Bri, Pri, Cri, W0r, W0i, W1r, W1i);
  s4_cauchy<<<(Llen * Nst + TB - 1) / TB, TB, 0, stream>>>(Lam, perl, APr, APi);
  s4_cauchy_gemm<<<dim3(Dch / 16, Llen / 16), 32, 0, stream>>>(
      APr, APi, W0r, W0i, W1r, W1i, perl, Ev);

  // --- K = ifft_L(evaluated): 16384 = 4*8^4 -> r4 pass then 4 r8 passes,
  //     ping-ponging Ev<->T0, 5 passes total, ends in T0 ------------------
  s4_fft_r4<<<gR4s, TB, 0, stream>>>(Ev, T0, Llen, 1, +1.0f);
  for (int p = 0; p < 4; ++p) {                         // Ns = 4*8^p
    const float2* src = (p & 1) ? Ev : T0;
    float2*       dst = (p & 1) ? T0 : Ev;
    s4_fft_r8<<<gR8s, TB, 0, stream>>>(src, dst, Llen, 4 << (3 * p), +1.0f);
  }
  // 5 passes: Ev->T0->Ev->T0->Ev->T0?  p0:T0->Ev, p1:Ev->T0, p2:T0->Ev, p3:Ev->T0
  // => final K time-domain data is in T0.

  // --- Kd = FFT_2L(pad(Re(K)/L)): fused pad+r8 then 4 generic r8 passes --
  s4_kpad_r8<<<gR8, TB, 0, stream>>>(T0, T1);
  for (int p = 0; p < 4; ++p) {                         // Ns = 8*8^p
    const float2* src = (p & 1) ? T0 : T1;
    float2*       dst = (p == 3) ? Kd : ((p & 1) ? T1 : T0);
    s4_fft_r8<<<gR8, TB, 0, stream>>>(src, dst, NFFT, 8 << (3 * p), -1.0f);
  }

  // --- per-batch FFT convolution -----------------------------------------
  for (int b = 0; b < Bsz; ++b) {
    const float* ub = u + (size_t)b * Llen * Dch;
    float*       ob = out + (size_t)b * Llen * Dch;
    // forward: fused pad+r8 (->T0), 4 generic r8 passes -> ends in T0
    s4_upad_r8<<<gR8, TB, 0, stream>>>(ub, T0);
    for (int p = 0; p < 4; ++p) {                       // Ns = 8*8^p
      const float2* src = (p & 1) ? T1 : T0;
      float2*       dst = (p & 1) ? T0 : T1;
      s4_fft_r8<<<gR8, TB, 0, stream>>>(src, dst, NFFT, 8 << (3 * p), -1.0f);
    }
    // inverse: fused cmul+r8 (T0*Kd -> T1), 3 generic r8 passes -> ends T0,
    // fused last r8 pass + epilogue
    s4_cmul_r8<<<gR8, TB, 0, stream>>>(T0, Kd, T1);
    for (int p = 0; p < 3; ++p) {                       // Ns = 8*8^p (8..512)
      const float2* src = (p & 1) ? T0 : T1;
      float2*       dst = (p & 1) ? T1 : T0;
      s4_fft_r8<<<gR8, TB, 0, stream>>>(src, dst, NFFT, 8 << (3 * p), +1.0f);
    }
    s4_final_r8<<<gR8, TB, 0, stream>>>(T0, ub, Dp, ob);
  }
}